// Pixelcoreg_Focalloss_twomodel_52312701665542
// MI455X (gfx1250) — compile-verified
//
#include <hip/hip_runtime.h>

// ---------------- problem constants (from reference) ----------------
#define BB 16
#define NN (512 * 512)          // 262144 pixels per batch row
#define BN (BB * NN)            // 4,194,304 total pixels
#define K_REMEMBER 209715       // int((1-0.2) * N)
#define NV4 (NN / 4)            // 65536 float4 groups per row
#define CHUNK 8192              // elements per TDM-staged chunk (32 KB)

typedef __attribute__((ext_vector_type(2)))  float v2f;
typedef __attribute__((ext_vector_type(8)))  float v8f;
typedef __attribute__((ext_vector_type(4)))  float f4;
typedef __attribute__((ext_vector_type(4)))  int   i4;
typedef __attribute__((ext_vector_type(4)))  unsigned u32x4;
typedef __attribute__((ext_vector_type(8)))  int   i32x8;
typedef __attribute__((ext_vector_type(4)))  int   i32x4;

// ---------------- WMMA wave-32 sum reduction ----------------
// A (16x4 f32): VGPR0 = lane partial, VGPR1 = 0; B = ones(4x16).
// D[m][*] = partial[m] + partial[m+16]; summing D's 8 VGPRs gives half-wave
// totals in lane groups [0,15]/[16,31]; shfl_xor(16) completes the sum.
__device__ __forceinline__ float wave_reduce_sum_wmma(float v) {
  v2f a; a[0] = v;    a[1] = 0.0f;
  v2f b; b[0] = 1.0f; b[1] = 1.0f;
  v8f c = {};
  v8f d = __builtin_amdgcn_wmma_f32_16x16x4_f32(false, a, false, b, (short)0, c,
                                                false, false);
  float s = d[0] + d[1] + d[2] + d[3] + d[4] + d[5] + d[6] + d[7];
  s += __shfl_xor(s, 16, 32);
  return s;
}

// float -> order-preserving uint key (ascending float == ascending key)
__device__ __forceinline__ unsigned fkey(float f) {
  unsigned u = __float_as_uint(f);
  return (u & 0x80000000u) ? ~u : (u | 0x80000000u);
}

// ---------------- TDM: 1-row tile of nElem f32 -> LDS ----------------
// D# group0: count=1, lds_addr, 57-bit global_addr, type=2 ("image").
// D# group1: data_size=4B, tensor_dim0=tile_dim0=nElem, tensor_dim1=tile_dim1=1,
//            tensor_dim0_stride=nElem, no padding/multicast/iterate.
__device__ __forceinline__ void tdm_load_row_f32(unsigned lds_off,
                                                 const float* gptr,
                                                 unsigned nElem) {
  unsigned long long ga = (unsigned long long)(const void*)gptr;
  u32x4 g0;
  g0[0] = 1u;                                              // count=1
  g0[1] = lds_off;                                         // lds_addr
  g0[2] = (unsigned)(ga & 0xFFFFFFFFu);                    // global_addr[31:0]
  g0[3] = (unsigned)((ga >> 32) & 0x1FFFFFFu) | (2u << 30); // addr[56:32]|type=2
  i32x8 g1;
  g1[0] = (int)(2u << 16);                                 // data_size = 4 bytes
  g1[1] = (int)((nElem & 0xFFFFu) << 16);                  // tensor_dim0[15:0]
  g1[2] = (int)((nElem >> 16) & 0xFFFFu) | (int)(1u << 16); // td0[31:16] | td1=1
  g1[3] = (int)((nElem & 0xFFFFu) << 16);                  // tile_dim0
  g1[4] = 1;                                               // tile_dim1 = 1
  g1[5] = (int)nElem;                                      // td0_stride[31:0]
  g1[6] = 0;                                               // stride hi / td1_stride
  g1[7] = 0;
  i32x4 z4 = {0, 0, 0, 0};
  i32x8 z8 = {0, 0, 0, 0, 0, 0, 0, 0};
  // 6-arg form (clang-23 / therock-10.0 headers): (g0, g1, g2, g3, g4, cpol)
  __builtin_amdgcn_tensor_load_to_lds(g0, g1, z4, z4, z8, 0);
}

// ---------------- kernels ----------------
__global__ void init_kernel(double* acc, unsigned* hist, unsigned* prefKey,
                            unsigned* kRem, unsigned* tieCnt) {
  int tid = blockIdx.x * blockDim.x + threadIdx.x;
  int stride = gridDim.x * blockDim.x;
  if (tid < 3) acc[tid] = 0.0;
  for (int i = tid; i < 4 * BB * 256; i += stride) hist[i] = 0u;
  if (tid < BB) {
    prefKey[tid] = 0u;
    kRem[tid]    = (unsigned)K_REMEMBER;
    tieCnt[tid]  = 0u;
  }
}

// One thread = 4 pixels (B128 loads). Fuses: loss compute + loss store +
// targets total (WMMA wave reduce) + radix pass-0 histogram (loss already
// in registers -> saves a full 16.8 MB sweep).
__global__ void __launch_bounds__(256)
fused_loss_kernel(const f4* __restrict__ in1, const f4* __restrict__ in2,
                  const i4* __restrict__ tgt, f4* __restrict__ lossBuf,
                  unsigned* __restrict__ hist, double* __restrict__ acc) {
  __shared__ unsigned sh[256];
  sh[threadIdx.x] = 0u;
  __syncthreads();

  const int g = blockIdx.x * 256 + threadIdx.x;   // float4 group, [0, BN/4)
  const int row  = g >> 16;                       // NV4 = 2^16 groups per row
  const int j    = g & (NV4 - 1);
  const int base = row * (2 * NV4) + j;

  f4 a0 = in1[base], a1 = in1[base + NV4];
  f4 c0 = in2[base], c1 = in2[base + NV4];
  i4 tv = tgt[g];

  f4 lo;
  float tAcc = 0.0f;
#pragma unroll
  for (int q = 0; q < 4; ++q) {
    float A0 = a0[q], A1 = a1[q], C0 = c0[q], C1 = c1[q];
    float t = (float)tv[q];
    // 2-class log-softmax, both models
    float m1  = fmaxf(A0, A1);
    float ls1 = m1 + logf(expf(A0 - m1) + expf(A1 - m1));
    float l10 = A0 - ls1, l11 = A1 - ls1;
    float p10 = expf(l10), p11 = expf(l11);
    float m2  = fmaxf(C0, C1);
    float ls2 = m2 + logf(expf(C0 - m2) + expf(C1 - m2));
    float l20 = C0 - ls2, l21 = C1 - ls2;
    float p20 = expf(l20), p21 = expf(l21);

    float om11 = 1.0f - p11, om10 = 1.0f - p10;
    float om21 = 1.0f - p21, om20 = 1.0f - p20;
    float f1 = -t * om11 * om11 * l11 - (1.0f - t) * om10 * om10 * l10;
    float f2 = -t * om21 * om21 * l21 - (1.0f - t) * om20 * om20 * l20;
    float k12 = p10 * (l10 - l20) + p11 * (l11 - l21);
    float k21 = p20 * (l20 - l10) + p21 * (l21 - l11);

    float loss = 0.5f * (f1 + f2) + 0.5f * (k12 + k21);
    lo[q] = loss;
    tAcc += t;
    atomicAdd(&sh[fkey(loss) >> 24], 1u);        // radix pass-0 bucket
  }
  lossBuf[g] = lo;

  float ws = wave_reduce_sum_wmma(tAcc);         // v_wmma_f32_16x16x4_f32
  if ((threadIdx.x & 31) == 0) atomicAdd(&acc[2], (double)ws);

  __syncthreads();
  unsigned cnt = sh[threadIdx.x];
  if (cnt) atomicAdd(&hist[row * 256 + threadIdx.x], cnt);  // PASS 0 slot
}

// Radix passes 1..3: TDM stages a 32 KB chunk of the loss row into LDS
// (tensor_load_to_lds + s_wait_tensorcnt), then 8 waves histogram from LDS.
template <int PASS>
__global__ void __launch_bounds__(256)
hist_tdm_kernel(const float* __restrict__ lossBuf,
                const unsigned* __restrict__ prefKey,
                unsigned* __restrict__ hist) {
  __shared__ float    buf[CHUNK];
  __shared__ unsigned sh[256];
  const int row   = blockIdx.x >> 5;             // 32 chunks per row
  const int chunk = blockIdx.x & 31;
  sh[threadIdx.x] = 0u;

  const float* src = lossBuf + row * NN + chunk * CHUNK;
  if (threadIdx.x < 32) {                        // wave 0 issues the DMA
    tdm_load_row_f32((unsigned)(unsigned long long)(void*)buf, src, CHUNK);
    __builtin_amdgcn_s_wait_tensorcnt(0);        // s_wait_tensorcnt 0x0
  }
  __syncthreads();

  const unsigned pref = prefKey[row];
  constexpr int HI = 32 - 8 * PASS;              // prefix-match shift
  constexpr int SH = 24 - 8 * PASS;              // current-byte shift
#pragma unroll 4
  for (int k = threadIdx.x; k < CHUNK; k += 256) {
    unsigned key = fkey(buf[k]);
    if ((key >> HI) == (pref >> HI))
      atomicAdd(&sh[(key >> SH) & 0xFFu], 1u);
  }
  __syncthreads();
  unsigned cnt = sh[threadIdx.x];
  if (cnt) atomicAdd(&hist[(PASS * BB + row) * 256 + threadIdx.x], cnt);
}

template <int PASS>
__global__ void scan_kernel(const unsigned* __restrict__ hist,
                            unsigned* prefKey, unsigned* kRem) {
  int row = threadIdx.x;
  if (row >= BB) return;
  const unsigned* h = hist + (PASS * BB + row) * 256;
  unsigned need = kRem[row];
  unsigned cum = 0, sel = 255;
  for (int v = 0; v < 256; ++v) {
    unsigned c = h[v];
    if (cum + c >= need) { sel = (unsigned)v; break; }
    cum += c;
  }
  prefKey[row] |= sel << (24 - 8 * PASS);
  kRem[row] = need - cum;     // after PASS 3: #kth-valued ties to include
}

// Selective sum: key < T fully included; key == T rationed by per-row atomic
// tie counter. B128 loads; WMMA wave reductions; f64 atomic accumulators.
__global__ void __launch_bounds__(256)
select_sum_kernel(const f4* __restrict__ lossBuf, const i4* __restrict__ tgt,
                  const unsigned* __restrict__ prefKey,
                  const unsigned* __restrict__ needTies,
                  unsigned* __restrict__ tieCnt, double* __restrict__ acc) {
  const int g   = blockIdx.x * 256 + threadIdx.x;   // [0, BN/4)
  const int row = g >> 16;
  f4 v  = lossBuf[g];
  i4 tv = tgt[g];
  const unsigned T  = prefKey[row];
  const unsigned nt = needTies[row];

  float lAcc = 0.0f, tAcc = 0.0f;
#pragma unroll
  for (int q = 0; q < 4; ++q) {
    unsigned key = fkey(v[q]);
    bool inc = (key < T);
    if (key == T) {
      unsigned pos = atomicAdd(&tieCnt[row], 1u);
      inc = (pos < nt);
    }
    if (inc) { lAcc += v[q]; tAcc += (float)tv[q]; }
  }
  float l = wave_reduce_sum_wmma(lAcc);          // v_wmma_f32_16x16x4_f32
  float t = wave_reduce_sum_wmma(tAcc);
  if ((threadIdx.x & 31) == 0) {
    atomicAdd(&acc[0], (double)l);
    atomicAdd(&acc[1], (double)t);
  }
}

__global__ void finalize_kernel(const double* __restrict__ acc,
                                float* __restrict__ out) {
  if (blockIdx.x == 0 && threadIdx.x == 0) {
    out[0] = (float)(acc[0] / ((double)BB * (double)K_REMEMBER)); // loss mean
    out[1] = (float)(acc[1] / acc[2]);                            // t_sel/t_sum
  }
}

// ---------------- host launch ----------------
extern "C" void kernel_launch(void* const* d_in, const int* in_sizes, int n_in,
                              void* d_out, int out_size, void* d_ws, size_t ws_size,
                              hipStream_t stream) {
  (void)in_sizes; (void)n_in; (void)out_size; (void)ws_size;
  const f4* in1 = (const f4*)d_in[0];   // [B,2,H,W] fp32
  const f4* in2 = (const f4*)d_in[1];   // [B,2,H,W] fp32
  const i4* tgt = (const i4*)d_in[2];   // [B,H,W] int32
  float* out = (float*)d_out;           // 2 scalars

  // workspace carve-up
  char* ws = (char*)d_ws;
  double*   acc     = (double*)ws;                       // 3 doubles (64 B)
  unsigned* hist    = (unsigned*)(ws + 64);              // 4*16*256 u32 = 64 KB
  unsigned* prefKey = (unsigned*)(ws + 64 + 4 * BB * 256 * 4);
  unsigned* kRem    = prefKey + BB;
  unsigned* tieCnt  = kRem + BB;
  f4*       lossBuf = (f4*)(ws + 65792);                 // 256-aligned, BN floats

  init_kernel<<<64, 256, 0, stream>>>(acc, hist, prefKey, kRem, tieCnt);

  fused_loss_kernel<<<BN / 4 / 256, 256, 0, stream>>>(in1, in2, tgt, lossBuf,
                                                      hist, acc);
  scan_kernel<0><<<1, 32, 0, stream>>>(hist, prefKey, kRem);

  hist_tdm_kernel<1><<<BB * 32, 256, 0, stream>>>((const float*)lossBuf, prefKey, hist);
  scan_kernel<1><<<1, 32, 0, stream>>>(hist, prefKey, kRem);
  hist_tdm_kernel<2><<<BB * 32, 256, 0, stream>>>((const float*)lossBuf, prefKey, hist);
  scan_kernel<2><<<1, 32, 0, stream>>>(hist, prefKey, kRem);
  hist_tdm_kernel<3><<<BB * 32, 256, 0, stream>>>((const float*)lossBuf, prefKey, hist);
  scan_kernel<3><<<1, 32, 0, stream>>>(hist, prefKey, kRem);

  select_sum_kernel<<<BN / 4 / 256, 256, 0, stream>>>(lossBuf, tgt, prefKey,
                                                      kRem, tieCnt, acc);
  finalize_kernel<<<1, 1, 0, stream>>>(acc, out);
}